// RbfHSIC_45423574122532
// MI455X (gfx1250) — compile-verified
//
#include <hip/hip_runtime.h>
#include <stdint.h>
#include <math.h>

#define N 4096
#define D 512
#define NBINS 65536

typedef __attribute__((ext_vector_type(2))) float v2f;
typedef __attribute__((ext_vector_type(8))) float v8f;

// ---------------- row squared norms: sq[i] = sum_k X[i,k]^2 ----------------
__global__ __launch_bounds__(128) void rownorm_k(const float* __restrict__ X,
                                                 float* __restrict__ sq) {
    __shared__ float red[128];
    const int row = blockIdx.x;
    const float* p = X + (size_t)row * D;
    const int t = threadIdx.x;
    float s = 0.f;
    for (int k = t; k < D; k += 128) { float v = p[k]; s += v * v; }
    red[t] = s; __syncthreads();
    for (int off = 64; off > 0; off >>= 1) {
        if (t < off) red[t] += red[t + off];
        __syncthreads();
    }
    if (t == 0) sq[row] = red[0];
}

// ---------------- Gram via f32 WMMA: r2[i,j] = max(sq_i + sq_j - 2*X_i.X_j, 0)
// Register-blocked: each wave computes one 16x64 strip (4 output tiles).
// A fragment loaded once per k-step, reused by 4 independent WMMA chains.
// f32 A layout: lane L (0-15): vgpr v = A[L][v] (K=0,1); lanes 16-31: K=2,3.
// For a Gram matrix the B tile (X^T) uses the identical per-lane addressing.
__global__ __launch_bounds__(256) void gram_k(const float* __restrict__ X,
                                              const float* __restrict__ sq,
                                              float* __restrict__ r2) {
    const int wave = threadIdx.x >> 5;
    const int lane = threadIdx.x & 31;
    const int t  = blockIdx.x * 8 + wave;      // 16384 wave-jobs (256 x 64)
    const int ti = t >> 6;                     // i-tile (16 rows)
    const int tg = t & 63;                     // j-group (64 cols)
    const int i0 = ti << 4, j0 = tg << 6;
    const int lm = lane & 15;                  // M (for A) / N (for B) within tile
    const int kh = (lane >> 4) << 1;           // K sub-offset: 0 or 2
    const float* pa  = X + (size_t)(i0 + lm) * D + kh;
    const float* pb0 = X + (size_t)(j0 + lm) * D + kh;
    const float* pb1 = pb0 + (size_t)16 * D;
    const float* pb2 = pb0 + (size_t)32 * D;
    const float* pb3 = pb0 + (size_t)48 * D;

    v8f acc0 = {0.f,0.f,0.f,0.f,0.f,0.f,0.f,0.f};
    v8f acc1 = acc0, acc2 = acc0, acc3 = acc0;
    for (int k = 0; k < D; k += 4) {
        const v2f a  = *(const v2f*)(pa  + k);
        const v2f b0 = *(const v2f*)(pb0 + k);
        const v2f b1 = *(const v2f*)(pb1 + k);
        const v2f b2 = *(const v2f*)(pb2 + k);
        const v2f b3 = *(const v2f*)(pb3 + k);
        acc0 = __builtin_amdgcn_wmma_f32_16x16x4_f32(false, a, false, b0, (short)0, acc0, false, false);
        acc1 = __builtin_amdgcn_wmma_f32_16x16x4_f32(false, a, false, b1, (short)0, acc1, false, false);
        acc2 = __builtin_amdgcn_wmma_f32_16x16x4_f32(false, a, false, b2, (short)0, acc2, false, false);
        acc3 = __builtin_amdgcn_wmma_f32_16x16x4_f32(false, a, false, b3, (short)0, acc3, false, false);
    }

    // C/D layout: vgpr v -> row v (lanes 0-15) or v+8 (lanes 16-31); col = lane%16
    const int mbase = i0 + ((lane >> 4) << 3);
    float si[8];
#pragma unroll
    for (int v = 0; v < 8; ++v) si[v] = sq[mbase + v];

    v8f accs[4] = {acc0, acc1, acc2, acc3};
#pragma unroll
    for (int q = 0; q < 4; ++q) {
        const int gj = j0 + (q << 4) + lm;
        const float sj = sq[gj];
#pragma unroll
        for (int v = 0; v < 8; ++v) {
            const int gi = mbase + v;
            float r = si[v] + sj - 2.0f * accs[q][v];
            r = fmaxf(r, 0.0f);
            if (gi == gj) r = 0.0f;            // exact zero diagonal (as reference)
            r2[(size_t)gi * N + gj] = r;
        }
    }
}

// ---------------- histogram helpers (integer atomics -> deterministic) -----
__global__ void zero_u32_k(uint32_t* __restrict__ p, int n) {
    int i = blockIdx.x * blockDim.x + threadIdx.x;
    if (i < n) p[i] = 0u;
}

__global__ void hist_hi_k(const float* __restrict__ r2, uint32_t* __restrict__ hist) {
    const size_t total  = (size_t)N * N;
    const size_t stride = (size_t)gridDim.x * blockDim.x;
    for (size_t i = (size_t)blockIdx.x * blockDim.x + threadIdx.x; i < total; i += stride) {
        const float r = r2[i];
        if (r > 0.0f) {
            const uint32_t bits = __float_as_uint(r);  // positive floats: monotone as u32
            atomicAdd(&hist[bits >> 16], 1u);
        }
    }
}

__global__ void hist_lo_k(const float* __restrict__ r2,
                          const uint32_t* __restrict__ sel,
                          uint32_t* __restrict__ hist) {
    const uint32_t bucket = sel[0];
    const size_t total  = (size_t)N * N;
    const size_t stride = (size_t)gridDim.x * blockDim.x;
    for (size_t i = (size_t)blockIdx.x * blockDim.x + threadIdx.x; i < total; i += stride) {
        const float r = r2[i];
        if (r > 0.0f) {
            const uint32_t bits = __float_as_uint(r);
            if ((bits >> 16) == bucket) atomicAdd(&hist[bits & 0xFFFFu], 1u);
        }
    }
}

// ---------------- rank selection over 65536 bins ---------------------------
// stage 0: rank = (total-1)/2 -> writes sel[0]=hi16 bucket, sel[1]=rank-in-bucket
// stage 1: rank = sel[1]      -> sigma bits = (sel[0]<<16)|lo16; writes 1/(2s+1e-8)
__global__ __launch_bounds__(256) void select_k(const uint32_t* __restrict__ hist,
                                                uint32_t* __restrict__ sel,
                                                float* __restrict__ invOut,
                                                int stage) {
    __shared__ uint32_t sums[256];
    __shared__ uint32_t pre[256];
    __shared__ uint32_t mrank;
    const int t = threadIdx.x;
    const int base = t * (NBINS / 256);
    uint32_t s = 0;
    for (int b = 0; b < NBINS / 256; ++b) s += hist[base + b];
    sums[t] = s;
    __syncthreads();
    if (t == 0) {
        uint32_t c = 0;
        for (int i = 0; i < 256; ++i) { pre[i] = c; c += sums[i]; }
        mrank = (stage == 0) ? ((c > 0u) ? (c - 1u) / 2u : 0u) : sel[1];
    }
    __syncthreads();
    const uint32_t m = mrank;
    uint32_t cum = pre[t];
    if (m >= cum && m < cum + sums[t]) {       // exactly one thread
        for (int b = 0; b < NBINS / 256; ++b) {
            const uint32_t cnt = hist[base + b];
            if (m < cum + cnt) {
                if (stage == 0) {
                    sel[0] = (uint32_t)(base + b);
                    sel[1] = m - cum;
                } else {
                    const uint32_t bits = (sel[0] << 16) | (uint32_t)(base + b);
                    const float sigma = __uint_as_float(bits);
                    invOut[0] = 1.0f / (2.0f * sigma + 1e-8f);
                }
                break;
            }
            cum += cnt;
        }
    }
}

// ---------------- K = exp(-r2*inv) in-place, plus row means ----------------
__global__ __launch_bounds__(256) void exprow_k(float* __restrict__ Km,
                                                const float* __restrict__ inv,
                                                float* __restrict__ mean) {
    __shared__ float red[256];
    const int row = blockIdx.x;
    const float iv = inv[0];
    float* p = Km + (size_t)row * N;
    float s = 0.f;
    for (int j = threadIdx.x; j < N; j += 256) {
        const float v = expf(-p[j] * iv);
        p[j] = v;
        s += v;
    }
    red[threadIdx.x] = s; __syncthreads();
    for (int off = 128; off > 0; off >>= 1) {
        if (threadIdx.x < off) red[threadIdx.x] += red[threadIdx.x + off];
        __syncthreads();
    }
    if (threadIdx.x == 0) mean[row] = red[0] * (1.0f / (float)N);
}

// ---------------- sum (Kx[i,j]-mx[i]) * (Ky[j,i]-my[j]), 64x64 tiles -------
__global__ __launch_bounds__(256) void hsic_k(const float* __restrict__ Kx,
                                              const float* __restrict__ Ky,
                                              const float* __restrict__ mx,
                                              const float* __restrict__ my,
                                              float* __restrict__ part) {
    __shared__ float tile[64][65];   // Ky tile, padded (conflict-free transpose read)
    __shared__ float red[256];
    const int tb = blockIdx.x;       // 4096 tiles
    const int i0 = (tb >> 6) << 6;
    const int j0 = (tb & 63) << 6;
    for (int idx = threadIdx.x; idx < 64 * 64; idx += 256) {
        const int r = idx >> 6, c = idx & 63;          // coalesced Ky rows
        tile[r][c] = Ky[(size_t)(j0 + r) * N + (i0 + c)];
    }
    __syncthreads();
    float acc = 0.f;
    for (int idx = threadIdx.x; idx < 64 * 64; idx += 256) {
        const int r = idx >> 6, c = idx & 63;          // r: i-offset, c: j-offset
        const float a = Kx[(size_t)(i0 + r) * N + (j0 + c)] - mx[i0 + r];
        const float b = tile[c][r] - my[j0 + c];       // Ky[j,i]
        acc += a * b;
    }
    red[threadIdx.x] = acc; __syncthreads();
    for (int off = 128; off > 0; off >>= 1) {
        if (threadIdx.x < off) red[threadIdx.x] += red[threadIdx.x + off];
        __syncthreads();
    }
    if (threadIdx.x == 0) part[tb] = red[0];
}

__global__ __launch_bounds__(256) void final_k(const float* __restrict__ part,
                                               float* __restrict__ out) {
    __shared__ float red[256];
    float s = 0.f;
    for (int i = threadIdx.x; i < 4096; i += 256) s += part[i];
    red[threadIdx.x] = s; __syncthreads();
    for (int off = 128; off > 0; off >>= 1) {
        if (threadIdx.x < off) red[threadIdx.x] += red[threadIdx.x + off];
        __syncthreads();
    }
    if (threadIdx.x == 0) out[0] = red[0] * (1.0f / (4095.0f * 4095.0f));
}

extern "C" void kernel_launch(void* const* d_in, const int* in_sizes, int n_in,
                              void* d_out, int out_size, void* d_ws, size_t ws_size,
                              hipStream_t stream) {
    (void)in_sizes; (void)n_in; (void)out_size; (void)ws_size;
    const float* X = (const float*)d_in[0];
    const float* Y = (const float*)d_in[1];
    float* out = (float*)d_out;

    // workspace layout (all f32/u32, ~128.6 MB)
    float*    r2x  = (float*)d_ws;                 // 64 MB, becomes Kx
    float*    r2y  = r2x + (size_t)N * N;          // 64 MB, becomes Ky
    float*    sqx  = r2y + (size_t)N * N;
    float*    sqy  = sqx + N;
    float*    mx   = sqy + N;
    float*    my   = mx + N;
    uint32_t* hist = (uint32_t*)(my + N);          // 65536 bins
    uint32_t* sel  = hist + NBINS;                 // [bucket, rank]
    float*    invv = (float*)(sel + 4);            // 1/(2*sigma+1e-8)
    float*    part = invv + 4;                     // 4096 block partials

    struct Mat { const float* src; float* r2; float* sq; float* mean; };
    const Mat mats[2] = { {X, r2x, sqx, mx}, {Y, r2y, sqy, my} };

    for (int m = 0; m < 2; ++m) {
        rownorm_k<<<N, 128, 0, stream>>>(mats[m].src, mats[m].sq);
        gram_k<<<2048, 256, 0, stream>>>(mats[m].src, mats[m].sq, mats[m].r2);
        // exact median of positive entries via 2-pass 16-bit radix select
        zero_u32_k<<<NBINS / 256, 256, 0, stream>>>(hist, NBINS);
        hist_hi_k<<<2048, 256, 0, stream>>>(mats[m].r2, hist);
        select_k<<<1, 256, 0, stream>>>(hist, sel, invv, 0);
        zero_u32_k<<<NBINS / 256, 256, 0, stream>>>(hist, NBINS);
        hist_lo_k<<<2048, 256, 0, stream>>>(mats[m].r2, sel, hist);
        select_k<<<1, 256, 0, stream>>>(hist, sel, invv, 1);
        exprow_k<<<N, 256, 0, stream>>>(mats[m].r2, invv, mats[m].mean);
    }
    hsic_k<<<4096, 256, 0, stream>>>(r2x, r2y, mx, my, part);
    final_k<<<1, 256, 0, stream>>>(part, out);
}